// PerWellMLPTransition_88347477279460
// MI455X (gfx1250) — compile-verified
//
#include <hip/hip_runtime.h>

// ---------------------------------------------------------------------------
// PerWellMLPTransition on MI455X (gfx1250), wave32 + V_WMMA_F32_16X16X4_F32.
// Pipeline: enc_kernel -> trans_kernel (fused At/Bt, never materialized)
//           -> heads_kernel -> inj_kernel
// All hot-loop loads are unconditional (clamped addresses + masks) so the
// compiler emits straight-line global_load_b64 + v_wmma streams, no EXEC
// branching.
// ---------------------------------------------------------------------------

typedef float v2f __attribute__((ext_vector_type(2)));
typedef float v8f __attribute__((ext_vector_type(8)));

#define NB   4096
#define LAT  128
#define UD   24
#define HIDN 200
#define ZUD  152

__device__ __forceinline__ v8f vzero8() {
  v8f v;
#pragma unroll
  for (int r = 0; r < 8; r++) v[r] = 0.f;
  return v;
}

// D = A(16x4 f32) * B(4x16 f32) + C(16x16 f32)
__device__ __forceinline__ v8f wmma4(v2f a, v2f b, v8f c) {
  return __builtin_amdgcn_wmma_f32_16x16x4_f32(false, a, false, b, (short)0, c,
                                               false, false);
}

// A fragment (16 rows x 4 k) from an LDS tile with row stride S.
// lane<16: M=lane,   K=k0+{0,1} ; lane>=16: M=lane-16, K=k0+2+{0,1}
__device__ __forceinline__ v2f afrag(const float* t, int S, int k0, int lane) {
  int half = lane >> 4, m = lane & 15;
  const float* p = t + m * S + k0 + 2 * half;
  v2f a;
  a.x = p[0];
  a.y = p[1];
  return a;
}

// unconditional aligned float2 load
__device__ __forceinline__ v2f ld2(const float* p) {
  return *(const v2f*)p;
}

// ---------------------------------------------------------------------------
// Kernel 1: encoder  hz = (relu(relu([zt,dt]We1^T+be1)We2^T+be2))We3^T+be3
// 64 threads = 2 waves, each wave owns a 16-row batch tile.
// N-padded columns (200..207) hold finite garbage that is never read back.
// ---------------------------------------------------------------------------
__global__ __launch_bounds__(64) void enc_kernel(
    const float* __restrict__ zt, const float* __restrict__ dtv,
    const float* __restrict__ We1, const float* __restrict__ be1,
    const float* __restrict__ We2, const float* __restrict__ be2,
    const float* __restrict__ We3, const float* __restrict__ be3,
    float* __restrict__ hz) {
  __shared__ float sm[2 * 2 * 16 * 208];  // per wave: X[16*208] + H[16*208]
  int lane = threadIdx.x & 31;
  int wid = threadIdx.x >> 5;
  int half = lane >> 4, nlo = lane & 15;
  int b0 = (blockIdx.x * 2 + wid) * 16;
  float* X = sm + wid * 2 * 16 * 208;
  float* H = X + 16 * 208;

  // X = [zt | dt | 0-pad] (cols 0..131 used; K padded to 132)
  for (int idx = lane; idx < 16 * 132; idx += 32) {
    int r = idx / 132, c = idx % 132;
    float v = (c < 128) ? zt[(b0 + r) * 128 + c]
                        : ((c == 128) ? dtv[b0 + r] : 0.f);
    X[r * 208 + c] = v;
  }
  __syncthreads();

  // GEMM1: H = relu(X @ We1^T + be1), N=200 (13 tiles), K=129
  // We1 rows are 129 floats (odd) -> component loads; K tail (k=128) peeled.
  for (int nt = 0; nt < 13; nt++) {
    int n = nt * 16 + nlo;
    int nc = (n < HIDN) ? n : (HIDN - 1);  // clamp; garbage cols unread
    const float* W = We1 + nc * 129;
    v8f acc = vzero8();
    for (int k0 = 0; k0 < 128; k0 += 4) {
      v2f a = afrag(X, 208, k0, lane);
      int kk = k0 + 2 * half;
      v2f b;
      b.x = W[kk];
      b.y = W[kk + 1];
      acc = wmma4(a, b, acc);
    }
    {  // K tail: only k=128 (dt column) is real
      v2f a = afrag(X, 208, 128, lane);
      float w128 = W[128];  // in-bounds for every row
      v2f b;
      b.x = half ? 0.f : w128;
      b.y = 0.f;
      acc = wmma4(a, b, acc);
    }
    float bias = be1[nc];
#pragma unroll
    for (int r = 0; r < 8; r++)
      H[(r + 8 * half) * 208 + nt * 16 + nlo] = fmaxf(acc[r] + bias, 0.f);
  }
  __syncthreads();

  // GEMM2: X = relu(H @ We2^T + be2), N=200, K=200
  for (int nt = 0; nt < 13; nt++) {
    int n = nt * 16 + nlo;
    int nc = (n < HIDN) ? n : (HIDN - 1);
    const float* W = We2 + nc * 200;
    v8f acc = vzero8();
    for (int k0 = 0; k0 < 200; k0 += 4) {
      v2f a = afrag(H, 208, k0, lane);
      v2f b = ld2(W + k0 + 2 * half);
      acc = wmma4(a, b, acc);
    }
    float bias = be2[nc];
#pragma unroll
    for (int r = 0; r < 8; r++)
      X[(r + 8 * half) * 208 + nt * 16 + nlo] = fmaxf(acc[r] + bias, 0.f);
  }
  __syncthreads();

  // GEMM3: hz = X @ We3^T + be3, N=128, K=200
  for (int nt = 0; nt < 8; nt++) {
    int n = nt * 16 + nlo;
    const float* W = We3 + n * 200;
    v8f acc = vzero8();
    for (int k0 = 0; k0 < 200; k0 += 4) {
      v2f a = afrag(X, 208, k0, lane);
      v2f b = ld2(W + k0 + 2 * half);
      acc = wmma4(a, b, acc);
    }
    float bias = be3[n];
#pragma unroll
    for (int r = 0; r < 8; r++)
      hz[(b0 + r + 8 * half) * 128 + n] = acc[r] + bias;
  }
}

// ---------------------------------------------------------------------------
// Kernel 2: fused transition. zt1[b,i] = (hz[b]@[WA_i|WB_i]^T + [bA|bB]) . v[b]
// where v = [zt | ut*dt] (len 152). At/Bt are never written to memory.
// 256 threads = 8 waves; block owns 32 batch rows (2 sub-tiles / wave);
// wave w handles i = w, w+8, ..., 120+w.
// A fragments for the hz tile are invariant across i and nt -> held in
// registers (32 K-steps x 2 sub-tiles) so the hot loop is 1 b64 load + 2 WMMA.
// ---------------------------------------------------------------------------
__global__ __launch_bounds__(256) void trans_kernel(
    const float* __restrict__ zt, const float* __restrict__ dtv,
    const float* __restrict__ ut, const float* __restrict__ hz,
    const float* __restrict__ WA, const float* __restrict__ bA,
    const float* __restrict__ WB, const float* __restrict__ bB,
    float* __restrict__ zt1) {
  __shared__ float hzT[32 * 132];  // hz tile, K padded (cols 128..131 = 0)
  __shared__ float vT[32 * 160];   // [zt | ut*dt | 0], stride 160
  int tid = threadIdx.x, lane = tid & 31, wid = tid >> 5;
  int half = lane >> 4, nlo = lane & 15;
  int b0 = blockIdx.x * 32;

  for (int idx = tid; idx < 32 * 132; idx += 256) {
    int r = idx / 132, c = idx % 132;
    hzT[idx] = (c < 128) ? hz[(b0 + r) * 128 + c] : 0.f;
  }
  for (int idx = tid; idx < 32 * 160; idx += 256) {
    int r = idx / 160, c = idx % 160;
    float v;
    if (c < 128)
      v = zt[(b0 + r) * 128 + c];
    else if (c < 152)
      v = ut[(b0 + r) * 24 + (c - 128)] * dtv[b0 + r];
    else
      v = 0.f;
    vT[idx] = v;
  }
  __syncthreads();

  // Preload all A fragments into registers (invariant over i and nt).
  v2f Ar[32][2];
#pragma unroll
  for (int ks = 0; ks < 32; ks++) {
    Ar[ks][0] = afrag(hzT, 132, ks * 4, lane);
    Ar[ks][1] = afrag(hzT + 16 * 132, 132, ks * 4, lane);
  }

  for (int i = wid; i < 128; i += 8) {
    float accs[2][8];
#pragma unroll
    for (int s = 0; s < 2; s++)
#pragma unroll
      for (int r = 0; r < 8; r++) accs[s][r] = 0.f;

    for (int nt = 0; nt < 10; nt++) {
      int jd = nt * 16 + nlo;  // column in [At_i | Bt_i], valid < 152
      const float* Wrow;
      float biasv;
      float maskf;
      if (jd < 128) {
        Wrow = WA + (size_t)(i * 128 + jd) * 128;
        biasv = bA[i * 128 + jd];
        maskf = 1.f;
      } else if (jd < ZUD) {
        Wrow = WB + (size_t)(i * 24 + (jd - 128)) * 128;
        biasv = bB[i * 24 + (jd - 128)];
        maskf = 1.f;
      } else {
        Wrow = WA + (size_t)(i * 128) * 128;  // clamped, finite garbage
        biasv = 0.f;
        maskf = 0.f;
      }
      const float* Wp = Wrow + 2 * half;
      v8f acc0 = vzero8();
      v8f acc1 = vzero8();
#pragma unroll 8
      for (int ks = 0; ks < 32; ks++) {
        v2f b = ld2(Wp + ks * 4);  // unconditional b64 load
        acc0 = wmma4(Ar[ks][0], b, acc0);
        acc1 = wmma4(Ar[ks][1], b, acc1);
      }
      // fuse: dot row of [At|Bt] against v = [zt | ut*dt]; masked for pad
#pragma unroll
      for (int r = 0; r < 8; r++) {
        float w0 = vT[(r + 8 * half) * 160 + jd] * maskf;
        float w1 = vT[(16 + r + 8 * half) * 160 + jd] * maskf;
        accs[0][r] += (acc0[r] + biasv) * w0;
        accs[1][r] += (acc1[r] + biasv) * w1;
      }
    }
    // reduce across the 16 lanes of each half-wave, write zt1[:, i]
#pragma unroll
    for (int s = 0; s < 2; s++)
#pragma unroll
      for (int r = 0; r < 8; r++) {
        float v = accs[s][r];
        v += __shfl_xor(v, 1, 32);
        v += __shfl_xor(v, 2, 32);
        v += __shfl_xor(v, 4, 32);
        v += __shfl_xor(v, 8, 32);
        if (nlo == 0)
          zt1[(b0 + s * 16 + r + 8 * half) * 128 + i] = v;
      }
  }
}

// ---------------------------------------------------------------------------
// Kernel 3: per-well heads. One wave per (16-row batch tile, head h).
// zu(152) -> 64 relu -> 64 relu -> 2, written into yt1[:, 2h:2h+2].
// ---------------------------------------------------------------------------
__global__ __launch_bounds__(32) void heads_kernel(
    const float* __restrict__ ut, const float* __restrict__ zt1,
    const float* __restrict__ Wh1, const float* __restrict__ bh1,
    const float* __restrict__ Wh2, const float* __restrict__ bh2,
    const float* __restrict__ Wh3, const float* __restrict__ bh3,
    float* __restrict__ yt1) {
  __shared__ float zu[16 * 160];
  __shared__ float g1[16 * 68];
  __shared__ float g2[16 * 68];
  int lane = threadIdx.x & 31;
  int half = lane >> 4, nlo = lane & 15;
  int b0 = blockIdx.x * 16;
  int h = blockIdx.y;

  for (int idx = lane; idx < 16 * 160; idx += 32) {
    int r = idx / 160, c = idx % 160;
    float v;
    if (c < 128)
      v = zt1[(b0 + r) * 128 + c];
    else if (c < 152)
      v = ut[(b0 + r) * 24 + (c - 128)];
    else
      v = 0.f;
    zu[idx] = v;
  }
  __syncthreads();

  const float* W1 = Wh1 + (size_t)h * 64 * 152;
  const float* W2 = Wh2 + (size_t)h * 64 * 64;
  const float* W3 = Wh3 + (size_t)h * 2 * 64;

  // layer 1: [16,152] x [152,64]
  for (int nt = 0; nt < 4; nt++) {
    int n = nt * 16 + nlo;
    const float* W = W1 + n * 152;
    v8f acc = vzero8();
    for (int k0 = 0; k0 < 152; k0 += 4) {
      v2f a = afrag(zu, 160, k0, lane);
      v2f b = ld2(W + k0 + 2 * half);
      acc = wmma4(a, b, acc);
    }
    float bias = bh1[h * 64 + n];
#pragma unroll
    for (int r = 0; r < 8; r++)
      g1[(r + 8 * half) * 68 + n] = fmaxf(acc[r] + bias, 0.f);
  }
  __syncthreads();

  // layer 2: [16,64] x [64,64]
  for (int nt = 0; nt < 4; nt++) {
    int n = nt * 16 + nlo;
    const float* W = W2 + n * 64;
    v8f acc = vzero8();
    for (int k0 = 0; k0 < 64; k0 += 4) {
      v2f a = afrag(g1, 68, k0, lane);
      v2f b = ld2(W + k0 + 2 * half);
      acc = wmma4(a, b, acc);
    }
    float bias = bh2[h * 64 + n];
#pragma unroll
    for (int r = 0; r < 8; r++)
      g2[(r + 8 * half) * 68 + n] = fmaxf(acc[r] + bias, 0.f);
  }
  __syncthreads();

  // layer 3: [16,64] x [64,2] (N padded to 16; cols 2..15 garbage, unwritten)
  {
    int nc = (nlo < 2) ? nlo : 1;  // clamped row
    const float* W = W3 + nc * 64;
    v8f acc = vzero8();
    for (int k0 = 0; k0 < 64; k0 += 4) {
      v2f a = afrag(g2, 68, k0, lane);
      v2f b = ld2(W + k0 + 2 * half);
      acc = wmma4(a, b, acc);
    }
    if (nlo < 2) {
      float bias = bh3[h * 2 + nlo];
#pragma unroll
      for (int r = 0; r < 8; r++)
        yt1[(b0 + r + 8 * half) * 40 + h * 2 + nlo] = acc[r] + bias;
    }
  }
}

// ---------------------------------------------------------------------------
// Kernel 4: injector head, inj = zu @ Wi^T + bi (tiny: 4096x8, K=152). VALU.
// ---------------------------------------------------------------------------
__global__ __launch_bounds__(256) void inj_kernel(
    const float* __restrict__ ut, const float* __restrict__ zt1,
    const float* __restrict__ Wi, const float* __restrict__ bi,
    float* __restrict__ yt1) {
  int t = blockIdx.x * blockDim.x + threadIdx.x;
  if (t >= NB * 8) return;
  int b = t >> 3, j = t & 7;
  const float* w = Wi + j * 152;
  float s = bi[j];
  for (int k = 0; k < 128; k++) s += zt1[b * 128 + k] * w[k];
  for (int u = 0; u < 24; u++) s += ut[b * 24 + u] * w[128 + u];
  yt1[b * 40 + 32 + j] = s;
}

// ---------------------------------------------------------------------------
extern "C" void kernel_launch(void* const* d_in, const int* in_sizes, int n_in,
                              void* d_out, int out_size, void* d_ws,
                              size_t ws_size, hipStream_t stream) {
  const float* zt = (const float*)d_in[0];
  const float* dt = (const float*)d_in[1];
  const float* ut = (const float*)d_in[2];
  const float* We1 = (const float*)d_in[3];
  const float* be1 = (const float*)d_in[4];
  const float* We2 = (const float*)d_in[5];
  const float* be2 = (const float*)d_in[6];
  const float* We3 = (const float*)d_in[7];
  const float* be3 = (const float*)d_in[8];
  const float* WA = (const float*)d_in[9];
  const float* bA = (const float*)d_in[10];
  const float* WB = (const float*)d_in[11];
  const float* bB = (const float*)d_in[12];
  const float* Wh1 = (const float*)d_in[13];
  const float* bh1 = (const float*)d_in[14];
  const float* Wh2 = (const float*)d_in[15];
  const float* bh2 = (const float*)d_in[16];
  const float* Wh3 = (const float*)d_in[17];
  const float* bh3 = (const float*)d_in[18];
  const float* Wi = (const float*)d_in[19];
  const float* bi = (const float*)d_in[20];

  float* out = (float*)d_out;
  float* zt1 = out;             // [B,128]
  float* yt1 = out + NB * LAT;  // [B,40]
  float* hz = (float*)d_ws;     // [B,128] scratch

  enc_kernel<<<NB / 32, 64, 0, stream>>>(zt, dt, We1, be1, We2, be2, We3, be3,
                                         hz);
  trans_kernel<<<NB / 32, 256, 0, stream>>>(zt, dt, ut, hz, WA, bA, WB, bB,
                                            zt1);
  heads_kernel<<<dim3(NB / 16, 16), 32, 0, stream>>>(ut, zt1, Wh1, bh1, Wh2,
                                                     bh2, Wh3, bh3, yt1);
  inj_kernel<<<(NB * 8 + 255) / 256, 256, 0, stream>>>(ut, zt1, Wi, bi, yt1);
}